// LightGCN_11081015623740
// MI455X (gfx1250) — compile-verified
//
#include <hip/hip_runtime.h>
#include <hip/hip_bf16.h>

typedef __attribute__((ext_vector_type(2))) float v2f;
typedef __attribute__((ext_vector_type(8))) float v8f;

#define NFEAT 128
#define NCLASS 32

// ---------------------------------------------------------------------------
// out[i*32 + c] = bias[c]   (projection accumulator starts at the bias)
// ---------------------------------------------------------------------------
__global__ __launch_bounds__(256) void init_out_kernel(float* __restrict__ out,
                                                       const float* __restrict__ bias,
                                                       int total) {
    int i = blockIdx.x * blockDim.x + threadIdx.x;
    if (i < total) out[i] = bias[i & (NCLASS - 1)];
}

// ---------------------------------------------------------------------------
// float4-vectorized zero of a workspace buffer (graph-capture safe memset)
// ---------------------------------------------------------------------------
__global__ __launch_bounds__(256) void zero_kernel(float4* __restrict__ p, int n4) {
    int i = blockIdx.x * blockDim.x + threadIdx.x;
    if (i < n4) p[i] = make_float4(0.f, 0.f, 0.f, 0.f);
}

// ---------------------------------------------------------------------------
// SpMM: hout[row[e]] += val[e] * hin[col[e]]
// One wave32 per edge. Each lane loads 16B (float4) of the 512B source row
// (32 lanes x 16B = 128 floats) and scatter-adds with global_atomic_add_f32.
// row/col/val are wave-uniform -> scalar loads. hin (51.2MB) fits in the
// 192MB L2, so the random gather and the atomics stay on-chip.
// ---------------------------------------------------------------------------
__global__ __launch_bounds__(256) void spmm_kernel(const int* __restrict__ erow,
                                                   const int* __restrict__ ecol,
                                                   const float* __restrict__ eval,
                                                   const float* __restrict__ hin,
                                                   float* __restrict__ hout,
                                                   int nEdges) {
    int gid  = blockIdx.x * blockDim.x + threadIdx.x;
    int edge = gid >> 5;
    int lane = threadIdx.x & 31;
    if (edge >= nEdges) return;

    int   r = erow[edge];
    int   c = ecol[edge];
    float v = eval[edge];

    const float4* src = (const float4*)(hin + (size_t)c * NFEAT);
    float4 d = src[lane];                       // global_load_b128

    float* dst = hout + (size_t)r * NFEAT + lane * 4;
    atomicAdd(dst + 0, v * d.x);
    atomicAdd(dst + 1, v * d.y);
    atomicAdd(dst + 2, v * d.z);
    atomicAdd(dst + 3, v * d.w);
}

// ---------------------------------------------------------------------------
// Projection with fp32 WMMA: out += scale * (h @ W)
// One wave computes a 16x32 output tile: two V_WMMA_F32_16X16X4_F32
// accumulators (N-tiles 0 and 1), K=128 walked in 32 steps of K=4.
//
// Layouts (ISA 7.12.2, wave32):
//   A 16x4 f32, 2 VGPRs: lane L holds row M=L&15; VGPR0/1 = K = 2*(L>>4)+{0,1}
//   B 4x16  f32, 2 VGPRs: lane L holds col N=L&15; VGPR0/1 = K = 2*(L>>4)+{0,1}
//   C/D 16x16 f32, 8 VGPRs: element [v + 8*(L>>4)][L&15] in VGPR v.
// N_NODES is a multiple of 16, so active waves always have EXEC all-ones;
// ragged tail waves exit as a whole before any WMMA.
// ---------------------------------------------------------------------------
__global__ __launch_bounds__(256) void proj_wmma_kernel(const float* __restrict__ h,
                                                        const float* __restrict__ W,
                                                        float* __restrict__ out,
                                                        float scale,
                                                        int nNodes) {
    int gid   = blockIdx.x * blockDim.x + threadIdx.x;
    int wave  = gid >> 5;
    int lane  = threadIdx.x & 31;
    int m0    = wave * 16;
    if (m0 >= nNodes) return;                   // whole-wave exit only

    int mrow  = lane & 15;                      // A row / B col / D col
    int khalf = lane >> 4;                      // K half select

    const float* hrow = h + (size_t)(m0 + mrow) * NFEAT;

    v8f c0 = {};
    v8f c1 = {};

    #pragma unroll 4
    for (int k0 = 0; k0 < NFEAT; k0 += 4) {
        int ka = k0 + khalf * 2;

        v2f a;
        a.x = hrow[ka];
        a.y = hrow[ka + 1];

        const float* wp = W + (size_t)ka * NCLASS + mrow;   // W is [128 x 32] row-major
        v2f b0, b1;
        b0.x = wp[0];            // K=ka,   N=mrow       (tile 0)
        b0.y = wp[NCLASS];       // K=ka+1, N=mrow
        b1.x = wp[16];           // K=ka,   N=mrow+16    (tile 1)
        b1.y = wp[NCLASS + 16];  // K=ka+1, N=mrow+16

        c0 = __builtin_amdgcn_wmma_f32_16x16x4_f32(false, a, false, b0,
                                                   (short)0, c0, false, false);
        c1 = __builtin_amdgcn_wmma_f32_16x16x4_f32(false, a, false, b1,
                                                   (short)0, c1, false, false);
    }

    // D layout -> global: element [v + 8*khalf][mrow]
    #pragma unroll
    for (int v = 0; v < 8; ++v) {
        size_t m = (size_t)(m0 + v + 8 * khalf);
        out[m * NCLASS + mrow]      += c0[v] * scale;   // each element owned by one lane
        out[m * NCLASS + mrow + 16] += c1[v] * scale;
    }
}

// ---------------------------------------------------------------------------
// Host-side launch: 3x (zero, SpMM, WMMA-projection), bias-initialized output.
// Workspace: two ping-pong [N,128] fp32 buffers (102.4 MB).
// ---------------------------------------------------------------------------
extern "C" void kernel_launch(void* const* d_in, const int* in_sizes, int n_in,
                              void* d_out, int out_size, void* d_ws, size_t ws_size,
                              hipStream_t stream) {
    const float* x    = (const float*)d_in[0];
    const int*   erow = (const int*)  d_in[1];
    const int*   ecol = (const int*)  d_in[2];
    const float* eval = (const float*)d_in[3];
    const float* W    = (const float*)d_in[4];
    const float* bias = (const float*)d_in[5];
    float*       out  = (float*)d_out;

    const int nNodes = in_sizes[0] / NFEAT;
    const int nEdges = in_sizes[1];

    float* buf0 = (float*)d_ws;
    float* buf1 = buf0 + (size_t)nNodes * NFEAT;

    const int   nFeatTotal = nNodes * NFEAT;        // 12.8M floats
    const int   n4         = nFeatTotal / 4;        // float4 count
    const int   zeroBlocks = (n4 + 255) / 256;
    const int   spmmBlocks = (nEdges * 32 + 255) / 256;
    const int   nTiles     = (nNodes + 15) / 16;
    const int   projBlocks = (nTiles * 32 + 255) / 256;
    const int   outTotal   = nNodes * NCLASS;
    const float third      = 1.0f / 3.0f;

    // out = bias broadcast
    init_out_kernel<<<(outTotal + 255) / 256, 256, 0, stream>>>(out, bias, outTotal);

    // layer 1: h1 = A x ; out += (h1 @ W)/3
    zero_kernel<<<zeroBlocks, 256, 0, stream>>>((float4*)buf0, n4);
    spmm_kernel<<<spmmBlocks, 256, 0, stream>>>(erow, ecol, eval, x, buf0, nEdges);
    proj_wmma_kernel<<<projBlocks, 256, 0, stream>>>(buf0, W, out, third, nNodes);

    // layer 2: h2 = A h1 ; out += (h2 @ W)/3
    zero_kernel<<<zeroBlocks, 256, 0, stream>>>((float4*)buf1, n4);
    spmm_kernel<<<spmmBlocks, 256, 0, stream>>>(erow, ecol, eval, buf0, buf1, nEdges);
    proj_wmma_kernel<<<projBlocks, 256, 0, stream>>>(buf1, W, out, third, nNodes);

    // layer 3: h3 = A h2 ; out += (h3 @ W)/3
    zero_kernel<<<zeroBlocks, 256, 0, stream>>>((float4*)buf0, n4);
    spmm_kernel<<<spmmBlocks, 256, 0, stream>>>(erow, ecol, eval, buf1, buf0, nEdges);
    proj_wmma_kernel<<<projBlocks, 256, 0, stream>>>(buf0, W, out, third, nNodes);
}